// DeformableConvBlock_15290083574326
// MI455X (gfx1250) — compile-verified
//
#include <hip/hip_runtime.h>

typedef __attribute__((ext_vector_type(2))) float v2f;
typedef __attribute__((ext_vector_type(4))) float v4f;
typedef __attribute__((ext_vector_type(8))) float v8f;

#define HH   128
#define WW   128
#define CIN  64
#define OCH  64
#define CK   576    // CIN * 9
#define NOFF 18     // 2*K*K offset channels
#define HW   16384  // H*W

// workspace layout (floats)
#define WS_BT   1179648            // after off[4][18][128][128]
#define WS_BTO  (1179648 + 36864)  // after Bt[576][64]

// ---------------------------------------------------------------------------
// Prep: transpose weight tables into k'-major order, k' = kk*64 + c.
//   Bt [k'][64] = weight[o][c][kk]          (coalesced B-fragment reads)
//   Bto[k'][32] = offw[n][c][kk], 0-padded for n >= 18
// ---------------------------------------------------------------------------
__global__ __launch_bounds__(64)
void prep_weights_kernel(const float* __restrict__ weight,
                         const float* __restrict__ offw,
                         float* __restrict__ Bt,
                         float* __restrict__ Bto) {
    const int k  = blockIdx.x;     // 0..575
    const int t  = threadIdx.x;    // 0..63
    const int c  = k & 63;
    const int kk = k >> 6;
    const int src = c * 9 + kk;
    Bt[k * 64 + t] = weight[t * CK + src];
    if (t < 32)
        Bto[k * 32 + t] = (t < NOFF) ? offw[t * CK + src] : 0.0f;
}

// ---------------------------------------------------------------------------
// Kernel 1: offset conv (3x3, SAME) via f32 WMMA implicit-im2col GEMM.
// K ordered kk-major: per-lane im2col base/mask per kk live in registers.
// ---------------------------------------------------------------------------
__global__ __launch_bounds__(256)
void offset_conv_kernel(const float* __restrict__ x,
                        const float* __restrict__ Bto,
                        const float* __restrict__ offb,
                        float* __restrict__ off) {
    const int lane  = threadIdx.x & 31;
    const int wave  = threadIdx.x >> 5;
    const int group = blockIdx.x * 8 + wave;   // 0..4095
    const int pbase = group * 16;
    const int b   = pbase >> 14;
    const int rem = pbase & (HW - 1);
    const int h   = rem >> 7;
    const int w0  = rem & (WW - 1);

    const float* xb = x + (size_t)b * CIN * HW;
    const int m     = lane & 15;
    const int khalf = (lane >> 4) << 1;        // ISA A/B lane layout
    const int wc    = w0 + m;

    int   base[9];
    float msk[9];
#pragma unroll
    for (int kk = 0; kk < 9; ++kk) {
        const int kh = kk / 3, kw = kk - kh * 3;
        const int iy = h - 1 + kh;
        const int ix = wc - 1 + kw;
        const int iyc = min(max(iy, 0), HH - 1);
        const int ixc = min(max(ix, 0), WW - 1);
        base[kk] = iyc * WW + ixc;
        msk[kk]  = (iy >= 0 && iy < HH && ix >= 0 && ix < WW) ? 1.0f : 0.0f;
    }

    v8f acc0 = {};
    v8f acc1 = {};
#pragma unroll
    for (int kk = 0; kk < 9; ++kk) {
        const float* xk = xb + base[kk];
        const float  mk = msk[kk];
        const float* bk = Bto + (size_t)(kk * 64) * 32;
        for (int cb = 0; cb < 64; cb += 4) {
            const int c0 = cb + khalf;
            v2f a;
            a.x = mk * xk[(size_t)c0 * HW];
            a.y = mk * xk[(size_t)(c0 + 1) * HW];
            const float* br0 = bk + (size_t)c0 * 32 + m;   // Bto[kk*64+c0][m]
            const float* br1 = br0 + 32;
            v2f b0 = {br0[0],  br1[0]};
            v2f b1 = {br0[16], br1[16]};
            acc0 = __builtin_amdgcn_wmma_f32_16x16x4_f32(false, a, false, b0,
                                                         (short)0, acc0, false, false);
            acc1 = __builtin_amdgcn_wmma_f32_16x16x4_f32(false, a, false, b1,
                                                         (short)0, acc1, false, false);
        }
    }

    // Epilogue: VGPR r -> M = r + 8*(lane>=16); N = lane&15. float4 stores.
    const int mbase = (lane >> 4) * 8;
    const int ncol  = lane & 15;
    {
        const float bv = offb[ncol];
        float* op = off + ((size_t)(b * NOFF + ncol)) * HW + h * WW + w0 + mbase;
        v4f lo = {acc0[0] + bv, acc0[1] + bv, acc0[2] + bv, acc0[3] + bv};
        v4f hi = {acc0[4] + bv, acc0[5] + bv, acc0[6] + bv, acc0[7] + bv};
        *(v4f*)(op)     = lo;
        *(v4f*)(op + 4) = hi;
    }
    const int n1 = 16 + ncol;
    if (n1 < NOFF) {
        const float bv = offb[n1];
        float* op = off + ((size_t)(b * NOFF + n1)) * HW + h * WW + w0 + mbase;
        v4f lo = {acc1[0] + bv, acc1[1] + bv, acc1[2] + bv, acc1[3] + bv};
        v4f hi = {acc1[4] + bv, acc1[5] + bv, acc1[6] + bv, acc1[7] + bv};
        *(v4f*)(op)     = lo;
        *(v4f*)(op + 4) = hi;
    }
}

// ---------------------------------------------------------------------------
// Kernel 2: deformable conv via f32 WMMA. One wave: 16 pixels x all 64 out
// channels (4 N-tiles share each A fragment). Per-lane bilinear parameters
// for all 9 taps live in registers (statically indexed: kk loop unrolled).
// ---------------------------------------------------------------------------
__global__ __launch_bounds__(256)
void deform_conv_kernel(const float* __restrict__ x,
                        const float* __restrict__ off,
                        const float* __restrict__ Bt,
                        const float* __restrict__ bias,
                        float* __restrict__ out) {
    const int lane  = threadIdx.x & 31;
    const int wave  = threadIdx.x >> 5;
    const int group = blockIdx.x * 8 + wave;
    const int pbase = group * 16;
    const int b   = pbase >> 14;
    const int rem = pbase & (HW - 1);
    const int h   = rem >> 7;
    const int w0  = rem & (WW - 1);

    const float* xb = x + (size_t)b * CIN * HW;
    const int mA    = lane & 15;
    const int khalf = (lane >> 4) << 1;
    const int ncol  = lane & 15;
    const int wpix  = w0 + mA;

    int   i00[9], i01[9], i10[9], i11[9];
    float f00[9], f01[9], f10[9], f11[9];
#pragma unroll
    for (int kk = 0; kk < 9; ++kk) {
        const int kh = kk / 3, kw = kk - kh * 3;
        const float* offp = off + (((size_t)b * NOFF + 2 * kk) * HH + h) * WW + wpix;
        const float oy = offp[0];
        const float ox = offp[HW];                 // channel 2*kk+1
        const float py = (float)(h - 1 + kh) + oy;
        const float px = (float)(wpix - 1 + kw) + ox;
        const float fy = floorf(py);
        const float fx = floorf(px);
        const float wy1 = py - fy, wx1 = px - fx;
        const float wy0 = 1.0f - wy1, wx0 = 1.0f - wx1;
        const int iy0 = (int)fy, ix0 = (int)fx;
        const int iy1 = iy0 + 1, ix1 = ix0 + 1;
        const float vy0 = (iy0 >= 0 && iy0 < HH) ? 1.0f : 0.0f;
        const float vy1 = (iy1 >= 0 && iy1 < HH) ? 1.0f : 0.0f;
        const float vx0 = (ix0 >= 0 && ix0 < WW) ? 1.0f : 0.0f;
        const float vx1 = (ix1 >= 0 && ix1 < WW) ? 1.0f : 0.0f;
        const int cy0 = min(max(iy0, 0), HH - 1);
        const int cy1 = min(max(iy1, 0), HH - 1);
        const int cx0 = min(max(ix0, 0), WW - 1);
        const int cx1 = min(max(ix1, 0), WW - 1);
        i00[kk] = cy0 * WW + cx0;
        i01[kk] = cy0 * WW + cx1;
        i10[kk] = cy1 * WW + cx0;
        i11[kk] = cy1 * WW + cx1;
        f00[kk] = wy0 * wx0 * vy0 * vx0;
        f01[kk] = wy0 * wx1 * vy0 * vx1;
        f10[kk] = wy1 * wx0 * vy1 * vx0;
        f11[kk] = wy1 * wx1 * vy1 * vx1;
    }

    v8f acc[4] = {v8f{}, v8f{}, v8f{}, v8f{}};
#pragma unroll
    for (int kk = 0; kk < 9; ++kk) {
        const int   a00 = i00[kk], a01 = i01[kk], a10 = i10[kk], a11 = i11[kk];
        const float g00 = f00[kk], g01 = f01[kk], g10 = f10[kk], g11 = f11[kk];
        const float* bk = Bt + (size_t)(kk * 64) * 64;
        if (kk < 8)   // prefetch next kk's 16KB weight block (global_prefetch_b8)
            __builtin_prefetch(Bt + (size_t)((kk + 1) * 64) * 64 + lane * 16, 0, 1);
        for (int cb = 0; cb < 64; cb += 4) {
            const int c0 = cb + khalf;
            const float* xc0 = xb + (size_t)c0 * HW;
            const float* xc1 = xc0 + HW;
            v2f a;
            a.x = g00 * xc0[a00] + g01 * xc0[a01] + g10 * xc0[a10] + g11 * xc0[a11];
            a.y = g00 * xc1[a00] + g01 * xc1[a01] + g10 * xc1[a10] + g11 * xc1[a11];
            const float* br0 = bk + (size_t)c0 * 64 + ncol;   // Bt[kk*64+c0][ncol]
            const float* br1 = br0 + 64;
            v2f b0 = {br0[0],  br1[0]};
            v2f b1 = {br0[16], br1[16]};
            v2f b2 = {br0[32], br1[32]};
            v2f b3 = {br0[48], br1[48]};
            acc[0] = __builtin_amdgcn_wmma_f32_16x16x4_f32(false, a, false, b0,
                                                           (short)0, acc[0], false, false);
            acc[1] = __builtin_amdgcn_wmma_f32_16x16x4_f32(false, a, false, b1,
                                                           (short)0, acc[1], false, false);
            acc[2] = __builtin_amdgcn_wmma_f32_16x16x4_f32(false, a, false, b2,
                                                           (short)0, acc[2], false, false);
            acc[3] = __builtin_amdgcn_wmma_f32_16x16x4_f32(false, a, false, b3,
                                                           (short)0, acc[3], false, false);
        }
    }

    // Epilogue: float4 stores (each lane owns 8 consecutive pixels per tile)
    const int mbase = (lane >> 4) * 8;
#pragma unroll
    for (int t = 0; t < 4; ++t) {
        const int o = t * 16 + ncol;
        const float bv = bias[o];
        float* op = out + (((size_t)b * OCH + o) * HH + h) * WW + w0 + mbase;
        v4f lo = {acc[t][0] + bv, acc[t][1] + bv, acc[t][2] + bv, acc[t][3] + bv};
        v4f hi = {acc[t][4] + bv, acc[t][5] + bv, acc[t][6] + bv, acc[t][7] + bv};
        *(v4f*)(op)     = lo;
        *(v4f*)(op + 4) = hi;
    }
}

// ---------------------------------------------------------------------------
extern "C" void kernel_launch(void* const* d_in, const int* in_sizes, int n_in,
                              void* d_out, int out_size, void* d_ws, size_t ws_size,
                              hipStream_t stream) {
    const float* x      = (const float*)d_in[0];
    const float* offw   = (const float*)d_in[1];
    const float* offb   = (const float*)d_in[2];
    const float* weight = (const float*)d_in[3];
    const float* bias   = (const float*)d_in[4];

    float* wsf = (float*)d_ws;          // needs 4,939,776 bytes total
    float* off = wsf;                   // [4][18][128][128]
    float* Bt  = wsf + WS_BT;           // [576][64]
    float* Bto = wsf + WS_BTO;          // [576][32], zero-padded

    prep_weights_kernel<<<576, 64, 0, stream>>>(weight, offw, Bt, Bto);
    offset_conv_kernel<<<512, 256, 0, stream>>>(x, Bto, offb, off);
    deform_conv_kernel<<<512, 256, 0, stream>>>(x, off, Bt, bias, (float*)d_out);
}